// LSTM_15135464751640
// MI455X (gfx1250) — compile-verified
//
#include <hip/hip_runtime.h>
#include <math.h>

// ---------------------------------------------------------------------------
// LSTM on MI455X (gfx1250): fused-gate recurrence via V_WMMA_F32_16X16X4_F32.
//   H=2048, D=1, C=10, B=256, T=256.
//   Per step: gates[4H,B] = Wh[4H,H] @ h[H,B] + Wx[4H,1] @ x_t[1,B] + b.
//   2.2 TFLOP fp32 total; Wh (64MB) is L2-resident (192MB L2).
//   v2: 64-col block tile -> each A fragment feeds 2 WMMAs (two 16-col
//   subtiles), halving L2 weight traffic (512MB -> 256MB per step) and
//   doubling the inner-loop WMMA:VMEM ratio.
// ---------------------------------------------------------------------------

#define H_DIM 2048
#define B_DIM 256
#define T_DIM 256
#define C_DIM 10
#define KC     64         // K-chunk staged in LDS per double-buffer phase
#define NCB    64         // batch columns per block
#define SPITCH 72         // LDS pitch (floats): 2*72 mod 64 == 16, so kq=0/1
                          // lane halves read disjoint bank sets at every
                          // 16-col subtile base (0/16/32/48)

typedef __attribute__((ext_vector_type(2))) float v2f;
typedef __attribute__((ext_vector_type(8))) float v8f;

__device__ __forceinline__ float sigmoidf_(float v) {
    return 1.0f / (1.0f + expf(-v));
}

// ---------------------------------------------------------------------------
// h0[m, b] = h_init[m] (broadcast over batch); c0 = c_init
// ---------------------------------------------------------------------------
__global__ __launch_bounds__(256)
void lstm_init_kernel(const float* __restrict__ h_init,
                      const float* __restrict__ c_init,
                      float* __restrict__ h0,
                      float* __restrict__ c0) {
    const int idx = blockIdx.x * 256 + threadIdx.x;   // over H*B
    const int m = idx >> 8;                           // row (B_DIM == 256)
    h0[idx] = h_init[m];
    c0[idx] = c_init[idx];
}

// ---------------------------------------------------------------------------
// One timestep. Block: 256 thr = 8 wave32 as 4 row-subtiles x 2 col-pairs
// -> 64-row x 64-col block tile. Each wave: 16 rows x 32 cols for ALL 4
// gates (4 gates x 2 col-subtiles = 8 f32 WMMA accumulators, 64 VGPRs).
// Grid: (H/64, B/64) = (32, 4).
// ---------------------------------------------------------------------------
__global__ __launch_bounds__(256, 2)
void lstm_step_kernel(const float* __restrict__ x,
                      const float* __restrict__ Wgh, const float* __restrict__ Wih,
                      const float* __restrict__ Wfh, const float* __restrict__ Woh,
                      const float* __restrict__ Wgx, const float* __restrict__ Wix,
                      const float* __restrict__ Wfx, const float* __restrict__ Wox,
                      const float* __restrict__ bg,  const float* __restrict__ bi,
                      const float* __restrict__ bf,  const float* __restrict__ bo,
                      const float* __restrict__ h_cur, const float* __restrict__ c_cur,
                      float* __restrict__ h_nxt, float* __restrict__ c_nxt,
                      int t) {
    // Double-buffered K-chunk of h (B-matrix), k-major, padded pitch.
    __shared__ float sh[2][KC][SPITCH];               // 36 KB

    const int tid  = threadIdx.x;
    const int lane = tid & 31;
    const int wave = tid >> 5;
    const int wr   = wave >> 1;          // 0..3 row subtile
    const int wc   = wave & 1;           // 0..1 col-pair subtile
    const int n    = lane & 15;          // fragment row/col index within tile
    const int kq   = lane >> 4;          // K quadrant (ISA 16x4 f32 layout)

    const int row0   = blockIdx.x * 64 + wr * 16;   // gate-row base (per gate)
    const int colblk = blockIdx.y * NCB;            // block's batch columns
    const int wcol   = wc * 32;                     // wave's 32-col window

    // acc[gate][col-subtile]
    v8f acc[4][2] = {};

    // Loader mapping: thread (lk, lcol) -> h[(k0+lk+4*i)*B + colblk+lcol]
    // (consecutive tids -> consecutive batch cols -> fully coalesced).
    const int lcol = tid & 63;
    const int lk   = tid >> 6;           // 0..3

    // Prologue: stage chunk 0.
    #pragma unroll
    for (int i = 0; i < KC / 4; ++i) {
        const int k = lk + i * 4;
        sh[0][k][lcol] = h_cur[(size_t)k * B_DIM + colblk + lcol];
    }
    __syncthreads();

    int buf = 0;
    for (int k0 = 0; k0 < H_DIM; k0 += KC) {
        // Prefetch next chunk into registers (overlaps with WMMA below).
        float pre[KC / 4];
        const int k0n = k0 + KC;
        if (k0n < H_DIM) {
            #pragma unroll
            for (int i = 0; i < KC / 4; ++i)
                pre[i] = h_cur[(size_t)(k0n + lk + i * 4) * B_DIM + colblk + lcol];
        }

        // WMMA over this chunk: K advances 4 per instruction; 4 gates x
        // 2 col-subtiles = 8 WMMA per 4 A-fragment loads (2:1 reuse).
        #pragma unroll
        for (int kk = 0; kk < KC; kk += 4) {
            const size_t kbase = (size_t)(k0 + kk + 2 * kq);
            // A fragments (16x4 f32): lane (16*kq + m) holds W[row0+m][kbase..+1]
            const v2f aG = *(const v2f*)(Wgh + (size_t)(row0 + n) * H_DIM + kbase);
            const v2f aI = *(const v2f*)(Wih + (size_t)(row0 + n) * H_DIM + kbase);
            const v2f aF = *(const v2f*)(Wfh + (size_t)(row0 + n) * H_DIM + kbase);
            const v2f aO = *(const v2f*)(Woh + (size_t)(row0 + n) * H_DIM + kbase);
            // B fragments (4x16 f32): lane (16*kq + n) holds h[kbase..+1][col+n]
            v2f b0, b1;
            b0.x = sh[buf][kk + 2 * kq    ][wcol + n];
            b0.y = sh[buf][kk + 2 * kq + 1][wcol + n];
            b1.x = sh[buf][kk + 2 * kq    ][wcol + 16 + n];
            b1.y = sh[buf][kk + 2 * kq + 1][wcol + 16 + n];

            acc[0][0] = __builtin_amdgcn_wmma_f32_16x16x4_f32(false, aG, false, b0,
                                                              (short)0, acc[0][0], false, false);
            acc[0][1] = __builtin_amdgcn_wmma_f32_16x16x4_f32(false, aG, false, b1,
                                                              (short)0, acc[0][1], false, false);
            acc[1][0] = __builtin_amdgcn_wmma_f32_16x16x4_f32(false, aI, false, b0,
                                                              (short)0, acc[1][0], false, false);
            acc[1][1] = __builtin_amdgcn_wmma_f32_16x16x4_f32(false, aI, false, b1,
                                                              (short)0, acc[1][1], false, false);
            acc[2][0] = __builtin_amdgcn_wmma_f32_16x16x4_f32(false, aF, false, b0,
                                                              (short)0, acc[2][0], false, false);
            acc[2][1] = __builtin_amdgcn_wmma_f32_16x16x4_f32(false, aF, false, b1,
                                                              (short)0, acc[2][1], false, false);
            acc[3][0] = __builtin_amdgcn_wmma_f32_16x16x4_f32(false, aO, false, b0,
                                                              (short)0, acc[3][0], false, false);
            acc[3][1] = __builtin_amdgcn_wmma_f32_16x16x4_f32(false, aO, false, b1,
                                                              (short)0, acc[3][1], false, false);
        }

        __syncthreads();
        if (k0n < H_DIM) {
            #pragma unroll
            for (int i = 0; i < KC / 4; ++i)
                sh[buf ^ 1][lk + i * 4][lcol] = pre[i];
        }
        __syncthreads();
        buf ^= 1;
    }

    // Epilogue: rank-1 x-term + bias, nonlinearities, state update.
    // D-matrix layout: VGPR r = row (row0 + r + 8*kq), col (colbase + n).
    #pragma unroll
    for (int cc = 0; cc < 2; ++cc) {
        const int col = colblk + wcol + cc * 16 + n;
        const float xv = x[(size_t)col * T_DIM + t];   // x is [B, T]
        #pragma unroll
        for (int r = 0; r < 8; ++r) {
            const int row = row0 + r + 8 * kq;
            const float g  = tanhf(acc[0][cc][r] + Wgx[row] * xv + bg[row]);
            const float ig = sigmoidf_(acc[1][cc][r] + Wix[row] * xv + bi[row]);
            const float fg = sigmoidf_(acc[2][cc][r] + Wfx[row] * xv + bf[row]);
            const float og = sigmoidf_(acc[3][cc][r] + Wox[row] * xv + bo[row]);
            const size_t off = (size_t)row * B_DIM + col;
            const float cn = g * ig + c_cur[off] * fg;
            c_nxt[off] = cn;
            h_nxt[off] = tanhf(cn) * og;
        }
    }
}

// ---------------------------------------------------------------------------
// Final projection: out[b, c] = sum_k Wph[c, k] * h[k, b] + bp[c].
// Tiny (10x2048x256); one block per class, one thread per batch column.
// ---------------------------------------------------------------------------
__global__ __launch_bounds__(256)
void lstm_proj_kernel(const float* __restrict__ Wph,
                      const float* __restrict__ bp,
                      const float* __restrict__ h,
                      float* __restrict__ out) {
    const int c = blockIdx.x;       // 0..C-1
    const int b = threadIdx.x;      // 0..B-1
    float s = bp[c];
    for (int k = 0; k < H_DIM; ++k)
        s += Wph[(size_t)c * H_DIM + k] * h[(size_t)k * B_DIM + b];
    out[(size_t)b * C_DIM + c] = s;
}

// ---------------------------------------------------------------------------
extern "C" void kernel_launch(void* const* d_in, const int* in_sizes, int n_in,
                              void* d_out, int out_size, void* d_ws, size_t ws_size,
                              hipStream_t stream) {
    (void)in_sizes; (void)n_in; (void)out_size; (void)ws_size;

    const float* x      = (const float*)d_in[0];
    const float* h_init = (const float*)d_in[1];
    const float* c_init = (const float*)d_in[2];
    const float* Wgx    = (const float*)d_in[3];
    const float* Wgh    = (const float*)d_in[4];
    const float* bg     = (const float*)d_in[5];
    const float* Wix    = (const float*)d_in[6];
    const float* Wih    = (const float*)d_in[7];
    const float* bi     = (const float*)d_in[8];
    const float* Wfx    = (const float*)d_in[9];
    const float* Wfh    = (const float*)d_in[10];
    const float* bf     = (const float*)d_in[11];
    const float* Wox    = (const float*)d_in[12];
    const float* Woh    = (const float*)d_in[13];
    const float* bo     = (const float*)d_in[14];
    const float* Wph    = (const float*)d_in[15];
    const float* bp     = (const float*)d_in[16];
    float* out = (float*)d_out;

    // Workspace: double-buffered h and c states, H*B floats each (2 MB each).
    const size_t HB = (size_t)H_DIM * B_DIM;
    float* hb[2] = { (float*)d_ws,           (float*)d_ws + HB };
    float* cb[2] = { (float*)d_ws + 2 * HB,  (float*)d_ws + 3 * HB };

    lstm_init_kernel<<<(H_DIM * B_DIM) / 256, 256, 0, stream>>>(h_init, c_init,
                                                                hb[0], cb[0]);

    int cur = 0;
    for (int t = 0; t < T_DIM; ++t) {
        lstm_step_kernel<<<dim3(H_DIM / 64, B_DIM / NCB), 256, 0, stream>>>(
            x, Wgh, Wih, Wfh, Woh, Wgx, Wix, Wfx, Wox,
            bg, bi, bf, bo,
            hb[cur], cb[cur], hb[cur ^ 1], cb[cur ^ 1], t);
        cur ^= 1;
    }

    lstm_proj_kernel<<<C_DIM, B_DIM, 0, stream>>>(Wph, bp, hb[cur], out);
}